// LogicDense_5523327943044
// MI455X (gfx1250) — compile-verified
//
#include <hip/hip_runtime.h>
#include <hip/hip_bf16.h>

typedef float v2f __attribute__((ext_vector_type(2)));
typedef float v4f __attribute__((ext_vector_type(4)));
typedef float v8f __attribute__((ext_vector_type(8)));

#define IN_DIM   8192
#define OUT_DIM  8192
#define BATCH    8192
#define RPB      4      // batch rows per workgroup (128 KB LDS stage)
#define TPB      256    // threads per block (8 waves)

// GATE_COEFFS, row-major 16x4 (all values in -2..2 -> packable as int8)
constexpr int GCI[16][4] = {
    {0, 0, 0, 0}, {0, 0, 0, 1}, {0, 1, 0, -1}, {0, 1, 0, 0},
    {0, 0, 1, -1}, {0, 0, 1, 0}, {0, 1, 1, -2}, {0, 1, 1, -1},
    {1, -1, -1, 1}, {1, -1, -1, 2}, {1, 0, -1, 0}, {1, 0, -1, 1},
    {1, -1, 0, 0}, {1, -1, 0, 1}, {1, 0, 0, -1}, {1, 0, 0, 0}};

// Pack rows rlo (byte idx 0..3) and rhi (byte idx 4..7) as signed bytes into
// one u64 immediate. Byte index at runtime = half*4 + nc.
constexpr unsigned long long packB(int rlo, int rhi) {
  unsigned long long p = 0;
  for (int i = 0; i < 4; ++i) {
    p |= (unsigned long long)((unsigned char)(signed char)GCI[rlo][i]) << (8 * i);
    p |= (unsigned long long)((unsigned char)(signed char)GCI[rhi][i]) << (8 * (i + 4));
  }
  return p;
}

__device__ __forceinline__ float gdecode(unsigned long long pack, int sh, bool act) {
  const int v = (int)(signed char)((unsigned)(pack >> sh) & 0xffu);
  return act ? (float)v : 0.0f;
}

// ---------------------------------------------------------------------------
// Kernel 1: c[m][0..3] = softmax(weight[m][:]) @ G   via V_WMMA_F32_16X16X4_F32
// One wave per 16-row block; K=16 as 4 chained K=4 WMMAs (all f32, no
// precision loss vs reference). Layouts per CDNA5 ISA §7.12.2:
//   A 16x4: VGPR0 = K0 (lanes 0-15) / K2 (lanes 16-31); VGPR1 = K1/K3
//   B 4x16: VGPR0 = row K0 / row K2; VGPR1 = row K1 / row K3 (N = lane&15)
//   D 16x16: VGPR r = row r (lanes 0-15) / row r+8 (lanes 16-31)
// ---------------------------------------------------------------------------
__global__ __launch_bounds__(32) void softgate_c_kernel(
    const float* __restrict__ weight, float* __restrict__ cout) {
  const int lane = threadIdx.x;        // 0..31
  const int half = lane >> 4;          // 0 or 1
  const int n    = lane & 15;          // D column / softmax row within tile
  const int m    = (int)blockIdx.x * 16 + n;

  const float4* wr4 = (const float4*)(weight + (size_t)m * 16);
  const float4 a0 = wr4[0], a1 = wr4[1], a2 = wr4[2], a3 = wr4[3];
  float wv[16] = {a0.x, a0.y, a0.z, a0.w, a1.x, a1.y, a1.z, a1.w,
                  a2.x, a2.y, a2.z, a2.w, a3.x, a3.y, a3.z, a3.w};
  float mx = wv[0];
#pragma unroll
  for (int k = 1; k < 16; ++k) mx = fmaxf(mx, wv[k]);
  float s = 0.f;
#pragma unroll
  for (int k = 0; k < 16; ++k) { wv[k] = expf(wv[k] - mx); s += wv[k]; }
  const float inv = 1.0f / s;
#pragma unroll
  for (int k = 0; k < 16; ++k) wv[k] *= inv;

  // Per-chunk packed B constants (q compile-time under full unroll).
  constexpr unsigned long long PX[4] = {packB(0, 2),  packB(4, 6),
                                        packB(8, 10), packB(12, 14)};
  constexpr unsigned long long PY[4] = {packB(1, 3),  packB(5, 7),
                                        packB(9, 11), packB(13, 15)};

  const bool act = (n < 4);
  const int  sh  = ((half << 2) | (n & 3)) * 8;  // byte index * 8
  v8f acc = {0.f, 0.f, 0.f, 0.f, 0.f, 0.f, 0.f, 0.f};
#pragma unroll
  for (int q = 0; q < 4; ++q) {
    // A: lane holds K = 4q+2*half, 4q+2*half+1 — constant wv indices, one
    // cndmask on `half` each (no dynamic register indexing).
    v2f A;
    A.x = half ? wv[4 * q + 2] : wv[4 * q + 0];
    A.y = half ? wv[4 * q + 3] : wv[4 * q + 1];
    // B: decoded from 64-bit integer immediates — pure VALU, no rodata,
    // no EXEC divergence around the WMMA chain.
    v2f B;
    B.x = gdecode(PX[q], sh, act);
    B.y = gdecode(PY[q], sh, act);
    acc = __builtin_amdgcn_wmma_f32_16x16x4_f32(
        /*neg_a=*/false, A, /*neg_b=*/false, B,
        /*c_mod=*/(short)0, acc, /*reuse_a=*/false, /*reuse_b=*/false);
  }

  if (act) {  // only first 4 D columns are real outputs
#pragma unroll
    for (int r = 0; r < 8; ++r) {
      const int M = half * 8 + r;
      cout[((size_t)blockIdx.x * 16 + M) * 4 + n] = acc[r];
    }
  }
}

// ---------------------------------------------------------------------------
// Kernel 2: the memory-bound gather/gate kernel (~512 MB HBM traffic total,
// floor ≈ 22 us at 23.3 TB/s). Stage RPB contiguous x rows (128 KB) into LDS
// via async global->LDS DMA, gather from LDS (ds_load), write output with
// non-temporal float4 stores so the 256 MB stream doesn't evict x/metadata
// from the 192 MB L2.
// ---------------------------------------------------------------------------
__global__ __launch_bounds__(TPB) void gate_kernel(
    const float* __restrict__ x,
    const int*   __restrict__ idx_a,
    const int*   __restrict__ idx_b,
    const float4* __restrict__ c,
    float* __restrict__ out) {
  __shared__ float smem[RPB * IN_DIM];  // 131072 bytes, offset 0 in LDS

  const int tid = threadIdx.x;
  const size_t row0 = (size_t)blockIdx.x * RPB;
  const float* src = x + row0 * IN_DIM;

  // Async DMA: RPB*IN_DIM floats = 8192 float4s, spread over TPB threads.
  // NOTE: smem is passed as an operand so it escapes — otherwise LLVM's
  // escape analysis assumes the asm can't write LDS and deletes the gathers.
#pragma unroll
  for (int k = 0; k < (RPB * IN_DIM) / (4 * TPB); ++k) {  // 32 iterations
    const int vec = k * TPB + tid;
    const unsigned ldsOff = (unsigned)(vec * 16);  // byte offset == LDS addr
    const unsigned long long gaddr = (unsigned long long)(src + vec * 4);
    asm volatile("global_load_async_to_lds_b128 %0, %1, off"
                 :: "v"(ldsOff), "v"(gaddr), "v"(smem)
                 : "memory");
  }
#if defined(__has_builtin) && __has_builtin(__builtin_amdgcn_s_wait_asynccnt)
  __builtin_amdgcn_s_wait_asynccnt(0);
#else
  asm volatile("s_wait_asynccnt 0" ::: "memory");
#endif
  __syncthreads();

  // 8192 columns / (256 threads * 4 cols) = 8 iterations.
#pragma unroll 2
  for (int k = 0; k < OUT_DIM / (TPB * 4); ++k) {
    const int j4 = (k * TPB + tid) * 4;
    const int4 ia = *(const int4*)(idx_a + j4);
    const int4 ib = *(const int4*)(idx_b + j4);
    const float4 c0 = c[j4 + 0];
    const float4 c1 = c[j4 + 1];
    const float4 c2 = c[j4 + 2];
    const float4 c3 = c[j4 + 3];
#pragma unroll
    for (int r = 0; r < RPB; ++r) {
      const float* sr = smem + r * IN_DIM;
      const float ax = sr[ia.x], bx = sr[ib.x];
      const float ay = sr[ia.y], by = sr[ib.y];
      const float az = sr[ia.z], bz = sr[ib.z];
      const float aw = sr[ia.w], bw = sr[ib.w];
      v4f o;
      o.x = c0.x + c0.y * ax + c0.z * bx + c0.w * (ax * bx);
      o.y = c1.x + c1.y * ay + c1.z * by + c1.w * (ay * by);
      o.z = c2.x + c2.y * az + c2.z * bz + c2.w * (az * bz);
      o.w = c3.x + c3.y * aw + c3.z * bw + c3.w * (aw * bw);
      __builtin_nontemporal_store(o, (v4f*)(out + (row0 + r) * OUT_DIM + j4));
    }
  }
}

extern "C" void kernel_launch(void* const* d_in, const int* in_sizes, int n_in,
                              void* d_out, int out_size, void* d_ws, size_t ws_size,
                              hipStream_t stream) {
  const float* x      = (const float*)d_in[0];
  const float* weight = (const float*)d_in[1];
  const int*   idx_a  = (const int*)d_in[2];
  const int*   idx_b  = (const int*)d_in[3];
  float* out = (float*)d_out;
  float* cws = (float*)d_ws;  // OUT_DIM x 4 floats = 128 KB scratch

  softgate_c_kernel<<<OUT_DIM / 16, 32, 0, stream>>>(weight, cws);
  gate_kernel<<<BATCH / RPB, TPB, 0, stream>>>(x, idx_a, idx_b,
                                               (const float4*)cws, out);
}